// SelfAttentionVarSizedElementReduce_41575283426056
// MI455X (gfx1250) — compile-verified
//
#include <hip/hip_runtime.h>
#include <math.h>

typedef __attribute__((ext_vector_type(2))) float v2f;
typedef __attribute__((ext_vector_type(8))) float v8f;

#define D128 128
#define LDSW 132   // LDS row stride (floats): pad 128->132 to break bank conflicts

// ---------------------------------------------------------------------------
// 1) segment starts: start[s] = first v with seg[v] == s; start[S] = V.
//    seg is sorted ascending. Thread v in [0, V] inclusive.
// ---------------------------------------------------------------------------
__global__ void starts_kernel(const int* __restrict__ seg, int* __restrict__ start,
                              int V, int S) {
  int v = blockIdx.x * blockDim.x + threadIdx.x;
  if (v > V) return;
  int prev = (v == 0) ? -1 : seg[v - 1];
  int cur  = (v == V) ? S  : seg[v];
  for (int s = prev + 1; s <= cur && s <= S; ++s) start[s] = v;
}

// ---------------------------------------------------------------------------
// 2) Cmat[d,d2] = sum_h Wk[h,d] * Wv[h,d2]   (128^3, tiny; fully cached)
//    kv[s] = meanE[s] @ Cmat  ==  (meanE @ Wk^T) @ Wv
// ---------------------------------------------------------------------------
__global__ void wkT_wv_kernel(const float* __restrict__ Wk,
                              const float* __restrict__ Wv,
                              float* __restrict__ C) {
  int d  = blockIdx.x;     // 128 blocks
  int d2 = threadIdx.x;    // 128 threads
  float acc = 0.f;
  for (int h = 0; h < D128; ++h)
    acc = fmaf(Wk[h * D128 + d], Wv[h * D128 + d2], acc);
  C[d * D128 + d2] = acc;
}

// ---------------------------------------------------------------------------
// 3) Fully fused attention: one block (4 waves) per 16 samples.
//    step 1: wave-per-sample segment mean of E        -> tA[16][128] (LDS)
//    step 2: kv tile = tA @ Cmat   (WMMA f32 16x16x4) -> tB[16][128] (LDS)
//    step 3: wave-per-sample online softmax + weighted segment sum of E
//            (single loop, flash-attention rescaling) -> tA (reused)
//    step 4: out tile = tA @ Wo^T  (WMMA)             -> d_out
//    E rows are read twice per sample; the second read hits L1/L2
//    (per-sample working set ~10 KB), so HBM sees E only once.
// ---------------------------------------------------------------------------
__global__ void fused_attn_kernel(const float* __restrict__ E,
                                  const int* __restrict__ start,
                                  const float* __restrict__ Cmat,
                                  const float* __restrict__ Wo,
                                  float* __restrict__ out, int S) {
  __shared__ float tA[16 * LDSW];
  __shared__ float tB[16 * LDSW];

  const int tid  = threadIdx.x;          // blockDim = 128 (4 waves)
  const int wave = tid >> 5;
  const int lane = tid & 31;
  const int half = lane >> 4;            // 0: lanes 0-15, 1: lanes 16-31
  const int l16  = lane & 15;
  const int s0   = blockIdx.x * 16;

  // ---- step 1: per-sample mean (wave per sample; lane = 4 columns) ----
  for (int j = 0; j < 4; ++j) {
    const int si = wave * 4 + j;
    const int s  = s0 + si;
    int b = 0, e = 0;
    if (s < S) { b = start[s]; e = start[s + 1]; }
    float4 acc = {0.f, 0.f, 0.f, 0.f};
    for (int i = b; i < e; ++i) {
      const float4 ev = ((const float4*)(E + (size_t)i * D128))[lane];
      acc.x += ev.x; acc.y += ev.y; acc.z += ev.z; acc.w += ev.w;
    }
    const float inv = 1.f / fmaxf((float)(e - b), 1.f);
    float4 mean4 = {acc.x * inv, acc.y * inv, acc.z * inv, acc.w * inv};
    ((float4*)(tA + si * LDSW))[lane] = mean4;   // row stride 132: 16B aligned
  }
  __syncthreads();

  // ---- step 2: tB = tA @ Cmat (each wave owns 2 of the 8 N-tiles) ----
  {
    v8f acc[2];
#pragma unroll
    for (int t = 0; t < 2; ++t) acc[t] = v8f{0.f,0.f,0.f,0.f,0.f,0.f,0.f,0.f};
#pragma unroll 4
    for (int k = 0; k < D128; k += 4) {
      v2f a;                                        // A 16x4: lanes0-15 K=k,k+1
      a.x = tA[l16 * LDSW + k + 2 * half];          //         lanes16-31 K=k+2,k+3
      a.y = tA[l16 * LDSW + k + 2 * half + 1];
#pragma unroll
      for (int t = 0; t < 2; ++t) {
        const int n0 = (wave * 2 + t) * 16;
        v2f bf;                                     // B[kk][n] = Cmat[k+kk][n0+n]
        bf.x = Cmat[(k + 2 * half) * D128 + n0 + l16];
        bf.y = Cmat[(k + 2 * half + 1) * D128 + n0 + l16];
        acc[t] = __builtin_amdgcn_wmma_f32_16x16x4_f32(
            false, a, false, bf, (short)0, acc[t], false, false);
      }
    }
#pragma unroll
    for (int t = 0; t < 2; ++t) {
      const int n0 = (wave * 2 + t) * 16;
#pragma unroll
      for (int j = 0; j < 8; ++j)                   // D: vgpr j -> row j+8*half
        tB[(j + 8 * half) * LDSW + n0 + l16] = acc[t][j];
    }
  }
  __syncthreads();

  // ---- step 3: online softmax + weighted sum (wave per sample) ----
  for (int j = 0; j < 4; ++j) {
    const int si = wave * 4 + j;
    const int s  = s0 + si;
    int b = 0, e = 0;
    if (s < S) { b = start[s]; e = start[s + 1]; }
    const float4 kv4 = ((const float4*)(tB + si * LDSW))[lane];
    float m = -INFINITY, l = 0.f;
    float4 acc = {0.f, 0.f, 0.f, 0.f};
    for (int i = b; i < e; ++i) {
      const float4 ev = ((const float4*)(E + (size_t)i * D128))[lane];
      float d = ev.x * kv4.x + ev.y * kv4.y + ev.z * kv4.z + ev.w * kv4.w;
#pragma unroll
      for (int off = 16; off; off >>= 1) d += __shfl_xor(d, off, 32);
      const float mn = fmaxf(m, d);              // all lanes hold full score
      const float sc = expf(m - mn);             // first iter: exp(-inf) = 0
      const float p  = expf(d - mn);
      l = l * sc + p;
      acc.x = acc.x * sc + p * ev.x;
      acc.y = acc.y * sc + p * ev.y;
      acc.z = acc.z * sc + p * ev.z;
      acc.w = acc.w * sc + p * ev.w;
      m = mn;
    }
    const float inv = (l > 0.f) ? 1.f / l : 0.f;
    float4 w4 = {acc.x * inv, acc.y * inv, acc.z * inv, acc.w * inv};
    ((float4*)(tA + si * LDSW))[lane] = w4;      // tA reused as wsum tile
  }
  __syncthreads();

  // ---- step 4: out = tA @ Wo^T (WMMA), store to global ----
  {
    v8f acc[2];
#pragma unroll
    for (int t = 0; t < 2; ++t) acc[t] = v8f{0.f,0.f,0.f,0.f,0.f,0.f,0.f,0.f};
#pragma unroll 4
    for (int k = 0; k < D128; k += 4) {
      v2f a;
      a.x = tA[l16 * LDSW + k + 2 * half];
      a.y = tA[l16 * LDSW + k + 2 * half + 1];
#pragma unroll
      for (int t = 0; t < 2; ++t) {
        const int n0 = (wave * 2 + t) * 16;
        v2f bf;                                   // B[kk][n] = Wo[n0+n][k+kk]
        bf.x = Wo[(size_t)(n0 + l16) * D128 + k + 2 * half];
        bf.y = Wo[(size_t)(n0 + l16) * D128 + k + 2 * half + 1];
        acc[t] = __builtin_amdgcn_wmma_f32_16x16x4_f32(
            false, a, false, bf, (short)0, acc[t], false, false);
      }
    }
#pragma unroll
    for (int t = 0; t < 2; ++t) {
      const int n0 = (wave * 2 + t) * 16;
#pragma unroll
      for (int j = 0; j < 8; ++j) {
        const int row = s0 + j + 8 * half;
        if (row < S) out[(size_t)row * D128 + n0 + l16] = acc[t][j];
      }
    }
  }
}

// ---------------------------------------------------------------------------
// launch
// ---------------------------------------------------------------------------
static inline size_t align256(size_t x) { return (x + 255) & ~(size_t)255; }

extern "C" void kernel_launch(void* const* d_in, const int* in_sizes, int n_in,
                              void* d_out, int out_size, void* d_ws, size_t ws_size,
                              hipStream_t stream) {
  const float* E   = (const float*)d_in[0];   // [V,128]
  const int*   seg = (const int*)d_in[1];     // [V] sorted
  const float* Wk  = (const float*)d_in[3];   // [128,128]
  const float* Wv  = (const float*)d_in[4];   // [128,128]
  const float* Wo  = (const float*)d_in[5];   // [128,128]
  float* out = (float*)d_out;                 // [S,128]

  const int V = in_sizes[0] / D128;
  const int S = out_size   / D128;            // S recovered from output shape

  char* ws = (char*)d_ws;
  size_t off = 0;
  int*   start = (int*)(ws + off);   off = align256(off + (size_t)(S + 1) * sizeof(int));
  float* Cmat  = (float*)(ws + off); off = align256(off + (size_t)D128 * D128 * sizeof(float));
  (void)ws_size; (void)n_in;

  starts_kernel<<<(V + 1 + 255) / 256, 256, 0, stream>>>(seg, start, V, S);
  wkT_wv_kernel<<<D128, D128, 0, stream>>>(Wk, Wv, Cmat);
  fused_attn_kernel<<<(S + 15) / 16, 128, 0, stream>>>(E, start, Cmat, Wo, out, S);
}